// GraphSAGEAggregator_31413390803231
// MI455X (gfx1250) — compile-verified
//
#include <hip/hip_runtime.h>

typedef float v2f __attribute__((ext_vector_type(2)));
typedef float v8f __attribute__((ext_vector_type(8)));

#define HIDDEN 128
#define LDS_PITCH 132  // pad 128 -> 132 floats: bank step 4 => conflict-free b64 LDS reads

// ---------------------------------------------------------------- zero scratch
__global__ __launch_bounds__(256) void sage_zero_kernel(float* __restrict__ p, int n) {
    int i = blockIdx.x * 256 + threadIdx.x;
    if (i < n) p[i] = 0.0f;
}

// ---------------------------------------------------------------- edge scatter
// One wave32 per edge: lane l moves float4 at cols [4l, 4l+4) -> 512B/edge, coalesced.
__global__ __launch_bounds__(256) void sage_scatter_kernel(
    const float* __restrict__ states, const int* __restrict__ esrc,
    const int* __restrict__ edst, float* __restrict__ nsum,
    float* __restrict__ cnt, int n_edges) {
    int edge = blockIdx.x * 8 + (threadIdx.x >> 5);
    if (edge >= n_edges) return;
    int lane = threadIdx.x & 31;
    int s = esrc[edge];
    int d = edst[edge];
    float4 v = ((const float4*)(states + (size_t)d * HIDDEN))[lane];
    float* o = nsum + (size_t)s * HIDDEN + lane * 4;
    atomicAdd(o + 0, v.x);
    atomicAdd(o + 1, v.y);
    atomicAdd(o + 2, v.z);
    atomicAdd(o + 3, v.w);
    if (lane == 0) atomicAdd(cnt + s, 1.0f);
}

// ---------------------------------------------------------------- mean normalize
__global__ __launch_bounds__(256) void sage_normalize_kernel(
    float* __restrict__ nsum, const float* __restrict__ cnt, int n_nodes) {
    int gid = blockIdx.x * 256 + threadIdx.x;
    int total = n_nodes * (HIDDEN / 4);
    if (gid >= total) return;
    int node = gid >> 5;
    int part = gid & 31;
    float c = cnt[node];
    float sc = (c > 0.0f) ? (1.0f / c) : 0.0f;
    float4* p = (float4*)(nsum + (size_t)node * HIDDEN) + part;
    float4 v = *p;
    v.x *= sc; v.y *= sc; v.z *= sc; v.w *= sc;
    *p = v;
}

// ---------------------------------------------------------------- fused dual GEMM + epilogue
// Block = 256 threads = 8 waves; block computes 16 node-rows x 128 cols.
// Wave w computes the 16x16 output tile at columns [16w, 16w+16) via
// V_WMMA_F32_16X16X4_F32 chains for both W_self and W_neigh.
__global__ __launch_bounds__(256) void sage_gemm_kernel(
    const float* __restrict__ states, const float* __restrict__ nmean,
    const float* __restrict__ cnt,
    const float* __restrict__ Wself, const float* __restrict__ bself,
    const float* __restrict__ Wneigh, const float* __restrict__ bneigh,
    float* __restrict__ out, int n_nodes) {
    __shared__ float lds_s[16 * LDS_PITCH];
    __shared__ float lds_n[16 * LDS_PITCH];

    int tid = threadIdx.x;
    int node_base = blockIdx.x * 16;

    // Cooperative stage of A tiles: 16 threads per row, 8 floats (2x float4) each.
    {
        int r = tid >> 4;
        int c = (tid & 15) * 8;
        int gr = node_base + r;
        if (gr >= n_nodes) gr = n_nodes - 1;  // clamp (grid is exact for N=100000)
        const float4* gs = (const float4*)(states + (size_t)gr * HIDDEN + c);
        const float4* gn = (const float4*)(nmean + (size_t)gr * HIDDEN + c);
        float4 s0 = gs[0], s1 = gs[1];
        float4 n0 = gn[0], n1 = gn[1];
        float4* ps = (float4*)&lds_s[r * LDS_PITCH + c];
        float4* pn = (float4*)&lds_n[r * LDS_PITCH + c];
        ps[0] = s0; ps[1] = s1;
        pn[0] = n0; pn[1] = n1;
    }
    __syncthreads();

    int wave  = tid >> 5;
    int lane  = tid & 31;
    int m     = lane & 15;          // A row / B col within tile
    int khalf = (lane >> 4) << 1;   // K sub-offset: lanes 0-15 -> K+0/K+1, 16-31 -> K+2/K+3
    int col   = wave * 16 + m;      // global output column (0..127)

    const float* as = &lds_s[m * LDS_PITCH + khalf];
    const float* an = &lds_n[m * LDS_PITCH + khalf];
    const float* bs = Wself  + (size_t)col * HIDDEN + khalf;  // B[k][col] = W[col][k]
    const float* bn = Wneigh + (size_t)col * HIDDEN + khalf;

    v8f cs = {0.f, 0.f, 0.f, 0.f, 0.f, 0.f, 0.f, 0.f};
    v8f cn = {0.f, 0.f, 0.f, 0.f, 0.f, 0.f, 0.f, 0.f};

#pragma unroll 8
    for (int k = 0; k < HIDDEN; k += 4) {
        v2f a0 = *(const v2f*)(as + k);
        v2f b0 = *(const v2f*)(bs + k);
        v2f a1 = *(const v2f*)(an + k);
        v2f b1 = *(const v2f*)(bn + k);
        cs = __builtin_amdgcn_wmma_f32_16x16x4_f32(false, a0, false, b0,
                                                   (short)0, cs, false, false);
        cn = __builtin_amdgcn_wmma_f32_16x16x4_f32(false, a1, false, b1,
                                                   (short)0, cn, false, false);
    }

    // Epilogue: C/D layout -> VGPR v, lane l: row M = v + (l>=16 ? 8 : 0), col N = l&15.
    float add_s = bself[col];
    float add_n = bneigh[col];
    int rbase = node_base + ((lane >> 4) << 3);
#pragma unroll
    for (int v = 0; v < 8; ++v) {
        int row = rbase + v;
        if (row < n_nodes) {
            float c   = cnt[row];
            float val = cs[v] + add_s;
            if (c > 0.0f) val += cn[v] + add_n;
            out[(size_t)row * HIDDEN + col] = fmaxf(val, 0.0f);
        }
    }
}

// ---------------------------------------------------------------- launch
extern "C" void kernel_launch(void* const* d_in, const int* in_sizes, int n_in,
                              void* d_out, int out_size, void* d_ws, size_t ws_size,
                              hipStream_t stream) {
    const float* states = (const float*)d_in[0];
    const int*   esrc   = (const int*)d_in[1];
    const int*   edst   = (const int*)d_in[2];
    const float* Wself  = (const float*)d_in[3];
    const float* bself  = (const float*)d_in[4];
    const float* Wneigh = (const float*)d_in[5];
    const float* bneigh = (const float*)d_in[6];
    float*       out    = (float*)d_out;

    int n_nodes = in_sizes[0] / HIDDEN;
    int n_edges = in_sizes[1];

    float* nsum = (float*)d_ws;
    float* cnt  = nsum + (size_t)n_nodes * HIDDEN;

    // 1) zero accumulators (nsum + cnt)
    int zero_n = n_nodes * (HIDDEN + 1);
    sage_zero_kernel<<<(zero_n + 255) / 256, 256, 0, stream>>>((float*)d_ws, zero_n);

    // 2) edge scatter-add (8 edges per 256-thread block, one wave32 per edge)
    sage_scatter_kernel<<<(n_edges + 7) / 8, 256, 0, stream>>>(
        states, esrc, edst, nsum, cnt, n_edges);

    // 3) normalize to mean
    int norm_n = n_nodes * (HIDDEN / 4);
    sage_normalize_kernel<<<(norm_n + 255) / 256, 256, 0, stream>>>(nsum, cnt, n_nodes);

    // 4) fused dual WMMA GEMM + bias + mask + ReLU
    sage_gemm_kernel<<<(n_nodes + 15) / 16, 256, 0, stream>>>(
        states, nsum, cnt, Wself, bself, Wneigh, bneigh, out, n_nodes);
}